// CantorGlobalAttention_14121852469495
// MI455X (gfx1250) — compile-verified
//
#include <hip/hip_runtime.h>
#include <hip/hip_bf16.h>

// ---------------- problem constants (match reference) ----------------
#define DIM     512
#define HEADS   8
#define HD      64          // DIM / HEADS
#define SEQ     2048
#define BATCH   2
#define KROUTE  64
#define ROWS    (BATCH * SEQ)      // 4096
#define NQKV    (3 * DIM)          // 1536

typedef float v2f __attribute__((ext_vector_type(2)));
typedef float v8f __attribute__((ext_vector_type(8)));

struct TileAcc {
  v8f c00, c01, c10, c11;   // 2x2 register block of 16x16 tiles
};

// -------------------------------------------------------------------
// 32x32 fp32 register-blocked tile: D = A(row-major, kdim) x W^T
// 4 x V_WMMA_F32_16X16X4_F32 per K-step of 4; 2 A frags + 2 B frags
// loaded per step -> 2x operand reuse vs the 1-tile version.
// A operand layout (32-bit 16x4): lanes 0-15 row M, VGPR0=K0,VGPR1=K1;
// lanes 16-31 row M, VGPR0=K2,VGPR1=K3.  B symmetric (N across lanes).
// -------------------------------------------------------------------
__device__ __forceinline__ TileAcc wmma_tile32_f32(const float* __restrict__ A,
                                                   const float* __restrict__ Wt,
                                                   int row0, int col0, int kdim) {
  const int lane = threadIdx.x & 31;
  const int mrow = lane & 15;
  const int kh   = (lane >> 4) << 1;              // 0 or 2
  const float* __restrict__ arow0 = A  + (size_t)(row0 + mrow) * kdim;
  const float* __restrict__ arow1 = arow0 + (size_t)16 * kdim;
  const float* __restrict__ brow0 = Wt + (size_t)(col0 + mrow) * kdim;
  const float* __restrict__ brow1 = brow0 + (size_t)16 * kdim;
  TileAcc t;
  t.c00 = (v8f){}; t.c01 = (v8f){}; t.c10 = (v8f){}; t.c11 = (v8f){};
  for (int kk = 0; kk < kdim; kk += 4) {
    const float2 a0f = *(const float2*)(arow0 + kk + kh);
    const float2 a1f = *(const float2*)(arow1 + kk + kh);
    const float2 b0f = *(const float2*)(brow0 + kk + kh);
    const float2 b1f = *(const float2*)(brow1 + kk + kh);
    v2f a0; a0.x = a0f.x; a0.y = a0f.y;
    v2f a1; a1.x = a1f.x; a1.y = a1f.y;
    v2f b0; b0.x = b0f.x; b0.y = b0f.y;
    v2f b1; b1.x = b1f.x; b1.y = b1f.y;
    t.c00 = __builtin_amdgcn_wmma_f32_16x16x4_f32(false, a0, false, b0, (short)0, t.c00, false, false);
    t.c01 = __builtin_amdgcn_wmma_f32_16x16x4_f32(false, a0, false, b1, (short)0, t.c01, false, false);
    t.c10 = __builtin_amdgcn_wmma_f32_16x16x4_f32(false, a1, false, b0, (short)0, t.c10, false, false);
    t.c11 = __builtin_amdgcn_wmma_f32_16x16x4_f32(false, a1, false, b1, (short)0, t.c11, false, false);
  }
  return t;
}

// -------------------------------------------------------------------
// Kernel 1 epilogue helper: scatter one 16x16 sub-tile of qkv into the
// (B, H, S, HD)-layout q/k/v buffers with bias add.
// -------------------------------------------------------------------
__device__ __forceinline__ void scatter_qkv_tile(const v8f& acc, int row0, int col0,
                                                 const float* __restrict__ b_qkv,
                                                 float* __restrict__ qb,
                                                 float* __restrict__ kb,
                                                 float* __restrict__ vb) {
  const int lane  = threadIdx.x & 31;
  const int n     = col0 + (lane & 15);
  const float bv  = b_qkv[n];
  const int which = n / DIM;                      // 0=q 1=k 2=v
  const int h     = (n % DIM) / HD;
  const int d     = n % HD;
  float* __restrict__ dst = (which == 0) ? qb : (which == 1) ? kb : vb;
  const int mbase = row0 + ((lane >> 4) << 3);
#pragma unroll
  for (int j = 0; j < 8; ++j) {
    const int m  = mbase + j;
    const int b_ = m >> 11;                       // / SEQ
    const int s  = m & (SEQ - 1);
    dst[(((size_t)(b_ * HEADS + h)) * SEQ + s) * HD + d] = acc[j] + bv;
  }
}

// -------------------------------------------------------------------
// Kernel 1: qkv = x @ w_qkv^T + b_qkv -> scattered q/k/v (B,H,S,HD).
// One wave per 32x32 output region.
// -------------------------------------------------------------------
__global__ void __launch_bounds__(256)
qkv_gemm_kernel(const float* __restrict__ x, const float* __restrict__ w_qkv,
                const float* __restrict__ b_qkv,
                float* __restrict__ qb, float* __restrict__ kb, float* __restrict__ vb) {
  const int wave = (blockIdx.x * blockDim.x + threadIdx.x) >> 5;
  const int TN   = NQKV / 32;                     // 48
  const int tm   = wave / TN;
  const int tn   = wave - tm * TN;
  const int row0 = tm * 32;
  const int col0 = tn * 32;

  TileAcc t = wmma_tile32_f32(x, w_qkv, row0, col0, DIM);

  scatter_qkv_tile(t.c00, row0,      col0,      b_qkv, qb, kb, vb);
  scatter_qkv_tile(t.c01, row0,      col0 + 16, b_qkv, qb, kb, vb);
  scatter_qkv_tile(t.c10, row0 + 16, col0,      b_qkv, qb, kb, vb);
  scatter_qkv_tile(t.c11, row0 + 16, col0 + 16, b_qkv, qb, kb, vb);
}

// -------------------------------------------------------------------
// Kernel 2: routed attention.  One wave per (b, h, s).  All k/v gathers
// are L2-resident (8 MB buffers << 192 MB L2).
// -------------------------------------------------------------------
__global__ void __launch_bounds__(256)
attn_kernel(const float* __restrict__ qb, const float* __restrict__ kb,
            const float* __restrict__ vb, const int* __restrict__ routes,
            float* __restrict__ att) {
  __shared__ __align__(16) float sq[8][HD];       // q row per wave
  __shared__ float sp[8][KROUTE];                 // probs per wave
  __shared__ int   sr[8][KROUTE];                 // clamped routes per wave

  const int wave = threadIdx.x >> 5;
  const int lane = threadIdx.x & 31;
  const int gw   = blockIdx.x * 8 + wave;         // [0, B*H*S)
  const int s    = gw & (SEQ - 1);
  const int bh   = gw >> 11;                      // b*HEADS + h

  // stage q row and clamped routes
  const float* __restrict__ qrow = qb + ((size_t)bh * SEQ + s) * HD;
  sq[wave][lane]      = qrow[lane];
  sq[wave][lane + 32] = qrow[lane + 32];
  {
    int r0 = routes[s * KROUTE + lane];
    int r1 = routes[s * KROUTE + lane + 32];
    r0 = r0 < 0 ? 0 : (r0 > SEQ - 1 ? SEQ - 1 : r0);
    r1 = r1 < 0 ? 0 : (r1 > SEQ - 1 ? SEQ - 1 : r1);
    sr[wave][lane]      = r0;
    sr[wave][lane + 32] = r1;
  }
  __syncthreads();

  // scores: 2 routed keys per lane, dot over HD=64 with float4 loads
  const float scale = 0.125f;                     // 1/sqrt(64)
  float sc[2];
#pragma unroll
  for (int jj = 0; jj < 2; ++jj) {
    const int j = lane + jj * 32;
    const int r = sr[wave][j];
    const float4* __restrict__ krow =
        (const float4*)(kb + ((size_t)bh * SEQ + r) * HD);
    float acc = 0.f;
#pragma unroll
    for (int d4 = 0; d4 < 16; ++d4) {
      const float4 kv = krow[d4];
      const float4 qv = *(const float4*)(&sq[wave][d4 * 4]);
      acc += kv.x * qv.x + kv.y * qv.y + kv.z * qv.z + kv.w * qv.w;
    }
    sc[jj] = acc * scale;
  }

  // wave32 softmax over the 64 scores (2 per lane)
  float m = fmaxf(sc[0], sc[1]);
#pragma unroll
  for (int off = 16; off >= 1; off >>= 1) m = fmaxf(m, __shfl_xor(m, off, 32));
  const float e0 = __expf(sc[0] - m);
  const float e1 = __expf(sc[1] - m);
  float sum = e0 + e1;
#pragma unroll
  for (int off = 16; off >= 1; off >>= 1) sum += __shfl_xor(sum, off, 32);
  const float inv = 1.0f / sum;
  sp[wave][lane]      = e0 * inv;
  sp[wave][lane + 32] = e1 * inv;
  __syncthreads();

  // out[d] = sum_j p_j * v[r_j][d];  2 d-values per lane
  float o0 = 0.f, o1 = 0.f;
#pragma unroll 4
  for (int j = 0; j < KROUTE; ++j) {
    const int r = sr[wave][j];
    const float p = sp[wave][j];
    const float* __restrict__ vrow = vb + ((size_t)bh * SEQ + r) * HD;
    o0 += p * vrow[lane];
    o1 += p * vrow[lane + 32];
  }

  // write in (B, S, D) layout: att[(b*SEQ+s)*DIM + h*HD + d]
  const int b_ = bh >> 3;
  const int h  = bh & 7;
  float* __restrict__ orow = att + ((size_t)(b_ * SEQ + s)) * DIM + h * HD;
  orow[lane]      = o0;
  orow[lane + 32] = o1;
}

// -------------------------------------------------------------------
// Kernel 3: out = att @ w_proj^T + b_proj -> d_out (B, S, D).
// One wave per 32x32 output region.
// -------------------------------------------------------------------
__device__ __forceinline__ void store_out_tile(const v8f& acc, int row0, int col0,
                                               const float* __restrict__ b_proj,
                                               float* __restrict__ out) {
  const int lane  = threadIdx.x & 31;
  const int n     = col0 + (lane & 15);
  const float bv  = b_proj[n];
  const int mbase = row0 + ((lane >> 4) << 3);
#pragma unroll
  for (int j = 0; j < 8; ++j) {
    out[(size_t)(mbase + j) * DIM + n] = acc[j] + bv;
  }
}

__global__ void __launch_bounds__(256)
proj_gemm_kernel(const float* __restrict__ att, const float* __restrict__ w_proj,
                 const float* __restrict__ b_proj, float* __restrict__ out) {
  const int wave = (blockIdx.x * blockDim.x + threadIdx.x) >> 5;
  const int TN   = DIM / 32;                      // 16
  const int tm   = wave / TN;
  const int tn   = wave - tm * TN;
  const int row0 = tm * 32;
  const int col0 = tn * 32;

  TileAcc t = wmma_tile32_f32(att, w_proj, row0, col0, DIM);

  store_out_tile(t.c00, row0,      col0,      b_proj, out);
  store_out_tile(t.c01, row0,      col0 + 16, b_proj, out);
  store_out_tile(t.c10, row0 + 16, col0,      b_proj, out);
  store_out_tile(t.c11, row0 + 16, col0 + 16, b_proj, out);
}

// -------------------------------------------------------------------
extern "C" void kernel_launch(void* const* d_in, const int* in_sizes, int n_in,
                              void* d_out, int out_size, void* d_ws, size_t ws_size,
                              hipStream_t stream) {
  const float* x      = (const float*)d_in[0];
  const int*   routes = (const int*)  d_in[1];
  const float* w_qkv  = (const float*)d_in[2];
  const float* b_qkv  = (const float*)d_in[3];
  const float* w_proj = (const float*)d_in[4];
  const float* b_proj = (const float*)d_in[5];
  float* out = (float*)d_out;

  const size_t per = (size_t)BATCH * HEADS * SEQ * HD;   // 2,097,152 floats
  float* qb  = (float*)d_ws;
  float* kb  = qb + per;
  float* vb  = kb + per;
  float* att = vb + per;

  // Kernel 1: (4096/32) x (1536/32) = 6144 waves, 8 waves/block -> 768 blocks
  {
    const int waves  = (ROWS / 32) * (NQKV / 32);
    const int blocks = waves / 8;
    qkv_gemm_kernel<<<blocks, 256, 0, stream>>>(x, w_qkv, b_qkv, qb, kb, vb);
  }
  // Kernel 2: B*H*S waves, 8 waves/block -> 4096 blocks
  {
    const int blocks = (BATCH * HEADS * SEQ) / 8;
    attn_kernel<<<blocks, 256, 0, stream>>>(qb, kb, vb, routes, att);
  }
  // Kernel 3: (4096/32) x (512/32) = 2048 waves -> 256 blocks
  {
    const int waves  = (ROWS / 32) * (DIM / 32);
    const int blocks = waves / 8;
    proj_gemm_kernel<<<blocks, 256, 0, stream>>>(att, w_proj, b_proj, out);
  }
}